// QuboGNN_82240033784024
// MI455X (gfx1250) — compile-verified
//
#include <hip/hip_runtime.h>
#include <hip/hip_bf16.h>

// ---------------------------------------------------------------------------
// 2-layer GCN on MI455X (gfx1250, wave32).
//  * GEMMs (50000x128 @ 128x128, fp32) run on the matrix pipe via the native
//    fp32 WMMA: v_wmma_f32_16x16x4_f32 (exact fp32 semantics, no downconvert).
//    W is staged in LDS as K-pairs (float2) so every B fragment is a single
//    aligned ds_load_b64 straight into an even VGPR pair -> no v_mov marshal.
//  * Edge aggregation (the roofline-dominant phase, ~800MB/layer of scattered
//    traffic) uses one wave per edge: coalesced float4 gather of h[src] and
//    hardware global_atomic_add_f32 scatter into agg[dst]. h and agg (25.6MB
//    each) both fit in the 192MB L2, so the random traffic stays on-die.
// ---------------------------------------------------------------------------

typedef __attribute__((ext_vector_type(2))) float v2f;
typedef __attribute__((ext_vector_type(8))) float v8f;

#define FEAT 128

// ---------------- degree / normalization ----------------
__global__ void deg_init_kernel(float* __restrict__ deg, int n) {
  int i = blockIdx.x * blockDim.x + threadIdx.x;
  if (i < n) deg[i] = 1.0f;  // self-loop contribution
}

__global__ void deg_edges_kernel(const long long* __restrict__ dst,
                                 float* __restrict__ deg, int E) {
  int i = blockIdx.x * blockDim.x + threadIdx.x;
  if (i < E)
    __hip_atomic_fetch_add(&deg[(int)dst[i]], 1.0f, __ATOMIC_RELAXED,
                           __HIP_MEMORY_SCOPE_AGENT);
}

__global__ void deg_rsqrt_kernel(float* __restrict__ deg, int n) {
  int i = blockIdx.x * blockDim.x + threadIdx.x;
  if (i < n) deg[i] = rsqrtf(deg[i]);  // deg >= 1 always (self loop)
}

// ---------------- fp32 WMMA GEMM: H[n x 128] = X[n x 128] * W[128 x 128] ----
// Block = 256 threads = 8 waves; block computes a 128-row slab, each wave a
// 16x128 stripe as 8 accumulators of 16x16, K stepped by 4 (32 iters, 256
// v_wmma per wave). LDS holds W packed as K-pairs:
//   sW2[p*128 + c] = { W[2p][c], W[2p+1][c] }   (p = 0..63, c = 0..127)
// so the B fragment for (kk, kk+1) x col is one contiguous 8B LDS load.
__global__ __launch_bounds__(256) void gemm_wmma_f32_kernel(
    const float* __restrict__ X, const float* __restrict__ W,
    float* __restrict__ H, int n) {
  __shared__ float2 sW2[(FEAT / 2) * FEAT];  // 64KB

  const int tid = threadIdx.x;
  // cooperative pack: thread i -> pair p = i>>7, col c = i&127 (coalesced per row)
  for (int i = tid; i < (FEAT / 2) * FEAT; i += 256) {
    const int p = i >> 7;
    const int c = i & (FEAT - 1);
    sW2[i] = make_float2(W[(2 * p) * FEAT + c], W[(2 * p + 1) * FEAT + c]);
  }
  __syncthreads();

  const int wave = tid >> 5;
  const int lane = tid & 31;
  const int laneLo = lane & 15;  // M (A) / N (B,D) position within half-wave
  const int laneHi = lane >> 4;  // selects K pair (A,B) / row group (D)
  const int rowBase = blockIdx.x * 128 + wave * 16;

  v8f acc[8];
#pragma unroll
  for (int t = 0; t < 8; ++t) acc[t] = (v8f){0, 0, 0, 0, 0, 0, 0, 0};

  // A-matrix 16x4 layout: lanes 0-15 hold K=k,k+1; lanes 16-31 hold K=k+2,k+3
  const int aRow = rowBase + laneLo;
  const int aRowC = (aRow < n) ? aRow : 0;  // clamp OOB reads (stores guarded)
  const float* __restrict__ xrow = X + (size_t)aRowC * FEAT;

  // per-lane LDS base: pair index advances by (k>>1), laneHi picks pair 0/1
  const v2f* __restrict__ sB = (const v2f*)sW2 + (size_t)laneHi * FEAT + laneLo;

  for (int k = 0; k < FEAT; k += 4) {
    const int kk = k + laneHi * 2;
    v2f a;
    a.x = xrow[kk];
    a.y = xrow[kk + 1];
    const v2f* __restrict__ sBk = sB + (size_t)(k >> 1) * FEAT;
#pragma unroll
    for (int t = 0; t < 8; ++t) {
      const v2f b = sBk[t * 16];  // single aligned ds_load_b64
      acc[t] = __builtin_amdgcn_wmma_f32_16x16x4_f32(
          false, a, false, b, (short)0, acc[t], false, false);
    }
  }

  // D layout: VGPR v -> row (rowBase + v + laneHi*8), col (t*16 + laneLo)
#pragma unroll
  for (int t = 0; t < 8; ++t) {
    const int col = t * 16 + laneLo;
#pragma unroll
    for (int v = 0; v < 8; ++v) {
      const int row = rowBase + v + laneHi * 8;
      if (row < n) H[(size_t)row * FEAT + col] = acc[t][v];
    }
  }
}

// ---------------- zero scratch ----------------
__global__ void zero_f32_kernel(float* __restrict__ p, size_t count) {
  size_t i = (size_t)blockIdx.x * blockDim.x + threadIdx.x;
  if (i < count) p[i] = 0.0f;
}

// ---------------- edge scatter: agg[dst] += h[src] * dinv[s]*dinv[d] --------
// One wave per edge; lane handles 4 features (float4 gather + 4 f32 atomics).
__global__ __launch_bounds__(256) void scatter_kernel(
    const long long* __restrict__ src, const long long* __restrict__ dst,
    const float* __restrict__ dinv, const float* __restrict__ h,
    float* __restrict__ agg, int E) {
  const int e = blockIdx.x * (blockDim.x >> 5) + (threadIdx.x >> 5);
  const int lane = threadIdx.x & 31;
  if (e >= E) return;
  // pull upcoming edge-index cachelines (global_prefetch_b8)
  if (e + 1024 < E) {
    __builtin_prefetch(src + e + 1024, 0, 0);
    __builtin_prefetch(dst + e + 1024, 0, 0);
  }
  const int s = (int)src[e];
  const int d = (int)dst[e];
  const float w = dinv[s] * dinv[d];
  const float4 v = ((const float4*)(h + (size_t)s * FEAT))[lane];
  float* ad = agg + (size_t)d * FEAT + lane * 4;
  __hip_atomic_fetch_add(ad + 0, v.x * w, __ATOMIC_RELAXED, __HIP_MEMORY_SCOPE_AGENT);
  __hip_atomic_fetch_add(ad + 1, v.y * w, __ATOMIC_RELAXED, __HIP_MEMORY_SCOPE_AGENT);
  __hip_atomic_fetch_add(ad + 2, v.z * w, __ATOMIC_RELAXED, __HIP_MEMORY_SCOPE_AGENT);
  __hip_atomic_fetch_add(ad + 3, v.w * w, __ATOMIC_RELAXED, __HIP_MEMORY_SCOPE_AGENT);
}

// ---------------- epilogue: agg = relu(agg + h*dinv^2 + bias) ---------------
// (self-loop term folded in here; agg becomes the next layer's input)
__global__ void epilogue_relu_kernel(float* __restrict__ agg,
                                     const float* __restrict__ h,
                                     const float* __restrict__ dinv,
                                     const float* __restrict__ bias,
                                     size_t nf) {
  size_t i = (size_t)blockIdx.x * blockDim.x + threadIdx.x;
  if (i >= nf) return;
  const int node = (int)(i >> 7);
  const int f = (int)(i & (FEAT - 1));
  const float di = dinv[node];
  const float v = agg[i] + h[i] * di * di + bias[f];
  agg[i] = v > 0.0f ? v : 0.0f;
}

// ---------------- column-sum partial reduction ----------------
__global__ void col_sum_kernel(const float* __restrict__ hmat,
                               float* __restrict__ csum, int n) {
  const int f = threadIdx.x;  // 128 threads = one column each
  float s = 0.0f;
  for (int i = blockIdx.x; i < n; i += gridDim.x) s += hmat[(size_t)i * FEAT + f];
  __hip_atomic_fetch_add(&csum[f], s, __ATOMIC_RELAXED, __HIP_MEMORY_SCOPE_AGENT);
}

// ---------------- final FC: out = mean(h2) @ Wfc + bfc ----------------
__global__ void fc_kernel(const float* __restrict__ csum,
                          const float* __restrict__ Wfc,
                          const float* __restrict__ bfc,
                          float* __restrict__ out, float invn) {
  const int c = threadIdx.x;
  if (c >= 32) return;
  float s = bfc[c];
  for (int f = 0; f < FEAT; ++f) s += (csum[f] * invn) * Wfc[f * 32 + c];
  out[c] = s;
}

// ---------------------------------------------------------------------------
extern "C" void kernel_launch(void* const* d_in, const int* in_sizes, int n_in,
                              void* d_out, int out_size, void* d_ws, size_t ws_size,
                              hipStream_t stream) {
  const float* x = (const float*)d_in[0];
  const long long* ei = (const long long*)d_in[1];  // int64 [2, E]
  const float* W1 = (const float*)d_in[2];
  const float* b1 = (const float*)d_in[3];
  const float* W2 = (const float*)d_in[4];
  const float* b2 = (const float*)d_in[5];
  const float* Wfc = (const float*)d_in[6];
  const float* bfc = (const float*)d_in[7];
  float* out = (float*)d_out;

  const int n = in_sizes[0] / FEAT;   // 50000
  const int E = in_sizes[1] / 2;      // 800000
  const long long* src = ei;
  const long long* dst = ei + E;

  // workspace layout (all 256B aligned):
  //   dinv  : n floats
  //   h     : n*128 floats
  //   agg   : n*128 floats
  //   csum  : 128 floats
  char* ws = (char*)d_ws;
  const size_t dinvBytes = (((size_t)n * 4) + 255) & ~(size_t)255;
  const size_t matBytes = (((size_t)n * FEAT * 4) + 255) & ~(size_t)255;
  float* dinv = (float*)ws;
  float* h = (float*)(ws + dinvBytes);
  float* agg = (float*)(ws + dinvBytes + matBytes);
  float* csum = (float*)(ws + dinvBytes + 2 * matBytes);

  const size_t nf = (size_t)n * FEAT;
  const int tb = 256;
  const int nBlk = (n + tb - 1) / tb;
  const int eBlk = (E + tb - 1) / tb;
  const int nfBlk = (int)((nf + tb - 1) / tb);
  const int gemmBlk = (n + 127) / 128;
  const int scatBlk = (E + 7) / 8;  // 8 waves (edges) per block

  // --- normalization: dinv = rsqrt(1 + in-degree) ---
  deg_init_kernel<<<nBlk, tb, 0, stream>>>(dinv, n);
  deg_edges_kernel<<<eBlk, tb, 0, stream>>>(dst, dinv, E);
  deg_rsqrt_kernel<<<nBlk, tb, 0, stream>>>(dinv, n);

  // --- layer 1 ---
  gemm_wmma_f32_kernel<<<gemmBlk, 256, 0, stream>>>(x, W1, h, n);
  zero_f32_kernel<<<nfBlk, tb, 0, stream>>>(agg, nf);
  scatter_kernel<<<scatBlk, 256, 0, stream>>>(src, dst, dinv, h, agg, E);
  epilogue_relu_kernel<<<nfBlk, tb, 0, stream>>>(agg, h, dinv, b1, nf);

  // --- layer 2 (agg now holds layer-2 input) ---
  gemm_wmma_f32_kernel<<<gemmBlk, 256, 0, stream>>>(agg, W2, h, n);
  zero_f32_kernel<<<nfBlk, tb, 0, stream>>>(agg, nf);
  scatter_kernel<<<scatBlk, 256, 0, stream>>>(src, dst, dinv, h, agg, E);
  epilogue_relu_kernel<<<nfBlk, tb, 0, stream>>>(agg, h, dinv, b2, nf);

  // --- mean pool + FC ---
  zero_f32_kernel<<<1, 128, 0, stream>>>(csum, 128);
  col_sum_kernel<<<512, 128, 0, stream>>>(agg, csum, n);
  fc_kernel<<<1, 32, 0, stream>>>(csum, Wfc, bfc, out, 1.0f / (float)n);
}